// ScaledDotProduct_721554506538
// MI455X (gfx1250) — compile-verified
//
#include <hip/hip_runtime.h>
#include <math.h>

typedef __attribute__((ext_vector_type(2))) float v2f;
typedef __attribute__((ext_vector_type(8))) float v8f;

#define D_FEAT 256

// ---------------------------------------------------------------- init
__global__ __launch_bounds__(256) void sdp_init(float* __restrict__ denom,
                                                float* __restrict__ gmax,
                                                int n_nodes) {
  int i = blockIdx.x * blockDim.x + threadIdx.x;
  if (i < n_nodes) denom[i] = 0.0f;
  if (i == 0) *gmax = -INFINITY;
}

__device__ inline void atomicMaxF(float* addr, float v) {
  unsigned int* ua = reinterpret_cast<unsigned int*>(addr);
  unsigned int cur = *ua;
  while (__uint_as_float(cur) < v) {
    unsigned int prev = atomicCAS(ua, cur, __float_as_uint(v));
    if (prev == cur) break;
    cur = prev;
  }
}

// ------------------------------------------------- pass A: WMMA scores
// One wave32 computes 16 edge dot-products via V_WMMA_F32_16X16X4_F32:
//  A[m][k] = q[row_{e_m}][K0+k]   (lane L holds row m=L%16, K-pair by half)
//  B[k][n] = k[col_{e_n}][K0+k]   (lane L holds col n=L%16, K-pair by half)
//  D[m][m] = score of edge m. 64 K-steps cover D_FEAT=256 (2 accumulators).
__global__ __launch_bounds__(256) void sdp_scores_wmma(
    const float* __restrict__ q, const float* __restrict__ kmat,
    const int* __restrict__ rows, const int* __restrict__ cols,
    float* __restrict__ scores, float* __restrict__ gmax, int E) {
  __shared__ float smax[8];
  const int lane = threadIdx.x & 31;
  const int wave = threadIdx.x >> 5;
  const int m    = lane & 15;
  const bool hi  = lane >= 16;
  const int edgeBase = (blockIdx.x * 8 + wave) * 16;

  int e  = edgeBase + m;
  int ec = e < E ? e : E - 1;            // clamp tail: keep EXEC all-ones
  const int r = rows[ec];
  const int c = cols[ec];
  const float* qp = q    + (size_t)r * D_FEAT + (hi ? 2 : 0);
  const float* kp = kmat + (size_t)c * D_FEAT + (hi ? 2 : 0);

  v8f acc0 = {};
  v8f acc1 = {};
#pragma unroll
  for (int ci = 0; ci < 64; ci += 2) {
    v2f a0 = *(const v2f*)(qp + 4 * ci);
    v2f b0 = *(const v2f*)(kp + 4 * ci);
    v2f a1 = *(const v2f*)(qp + 4 * ci + 4);
    v2f b1 = *(const v2f*)(kp + 4 * ci + 4);
    acc0 = __builtin_amdgcn_wmma_f32_16x16x4_f32(false, a0, false, b0,
                                                 (short)0, acc0, false, false);
    acc1 = __builtin_amdgcn_wmma_f32_16x16x4_f32(false, a1, false, b1,
                                                 (short)0, acc1, false, false);
  }

  // Diagonal of D: edge m<8 -> lane m, comp m; edge m>=8 -> lane m+16, comp m-8
  const int comp = lane & 7;
  float s = acc0[comp] + acc1[comp];
  const bool active = (lane < 8) || (lane >= 24);
  const int myE = edgeBase + ((lane < 8) ? lane : (lane - 16));
  float sc = s * (1.0f / 16.0f);         // TEMPERATURE = 16
  float red = (active && myE < E) ? sc : -INFINITY;
  if (active && myE < E) scores[myE] = sc;

  // wave -> block max reduction, one atomic per block
#pragma unroll
  for (int off = 16; off > 0; off >>= 1)
    red = fmaxf(red, __shfl_xor(red, off, 32));
  if (lane == 0) smax[wave] = red;
  __syncthreads();
  if (threadIdx.x == 0) {
    float mx = smax[0];
#pragma unroll
    for (int i = 1; i < 8; ++i) mx = fmaxf(mx, smax[i]);
    atomicMaxF(gmax, mx);
  }
}

// --------------------------------------- pass B: exp + per-row denom sum
__global__ __launch_bounds__(256) void sdp_exp_denom(
    const int* __restrict__ rows, float* __restrict__ scores,
    float* __restrict__ denom, const float* __restrict__ gmax, int E) {
  int e = blockIdx.x * blockDim.x + threadIdx.x;
  if (e < E) {
    float v = expf(scores[e] - *gmax);
    scores[e] = v;
    atomicAdd(&denom[rows[e]], v);       // global_atomic_add_f32
  }
}

// ------------------------------------------------- pass C: normalize
__global__ __launch_bounds__(256) void sdp_normalize(
    const int* __restrict__ rows, const float* __restrict__ scores,
    const float* __restrict__ denom, float* __restrict__ out, int E) {
  int e = blockIdx.x * blockDim.x + threadIdx.x;
  if (e < E) out[e] = scores[e] / denom[rows[e]];
}

extern "C" void kernel_launch(void* const* d_in, const int* in_sizes, int n_in,
                              void* d_out, int out_size, void* d_ws, size_t ws_size,
                              hipStream_t stream) {
  (void)n_in; (void)ws_size;
  const float* q   = (const float*)d_in[0];
  const float* k   = (const float*)d_in[1];
  const int*   idx = (const int*)d_in[2];   // [2, E] int32 (JAX x64 disabled)

  const int E       = out_size;             // one output per edge
  const int n_nodes = in_sizes[0] / D_FEAT; // q is [n_nodes, 256]
  const int* rows = idx;
  const int* cols = idx + E;

  float* ws     = (float*)d_ws;
  float* scores = ws;                 // E floats
  float* denom  = ws + E;             // n_nodes floats
  float* gmax   = ws + E + n_nodes;   // 1 float

  sdp_init<<<(n_nodes + 255) / 256, 256, 0, stream>>>(denom, gmax, n_nodes);

  int blocksA = (E + 127) / 128;      // 8 waves/block * 16 edges/wave
  sdp_scores_wmma<<<blocksA, 256, 0, stream>>>(q, k, rows, cols, scores, gmax, E);

  int blocksE = (E + 255) / 256;
  sdp_exp_denom<<<blocksE, 256, 0, stream>>>(rows, scores, denom, gmax, E);
  sdp_normalize<<<blocksE, 256, 0, stream>>>(rows, scores, denom, (float*)d_out, E);
}